// DepthDeformableTransformerEncoderLayer_41446434407089
// MI455X (gfx1250) — compile-verified
//
#include <hip/hip_runtime.h>
#include <hip/hip_bf16.h>
#include <math.h>

// ---------------------------------------------------------------------------
// DepthDeformableTransformerEncoderLayer for MI455X (gfx1250, wave32, WMMA)
//
// fp32 end-to-end: the layer is bandwidth/gather bound (21 GFLOP of GEMM vs
// ~300MB traffic @ 23.3 TB/s), and the bilinear sample coordinates are
// numerically sensitive, so V_WMMA_F32_16X16X4_F32 is the right matrix path.
// ---------------------------------------------------------------------------

typedef __attribute__((ext_vector_type(2))) float v2f;
typedef __attribute__((ext_vector_type(8))) float v8f;

#define ROWS_   30000        // BATCH * LQ  (== BATCH * LS here)
#define LQ_     15000
#define LS_     15000
#define CDIM_   256
#define HLVL_   100
#define WLVL_   150

enum { EPI_NONE = 0, EPI_GELU = 1 };

// ---------------------------------------------------------------------------
// WMMA fp32 GEMM:  C[M,N] = A[M,K] @ B[K,N] + bias   (row major)
// block = 256 thr = 8 waves; block tile 128(M) x 64(N); wave tile 16 x 64.
// B panel staged through LDS in K-chunks of 32 (32x65 fp32, pad = no bank
// conflicts).  A fragment per the CDNA5 16x4 fp32 layout:
//   lanes 0-15 : M = lane,    VGPR0/1 = K,K+1
//   lanes 16-31: M = lane-16, VGPR0/1 = K+2,K+3
// ---------------------------------------------------------------------------
template<int EPI, bool USE_MASK>
__global__ __launch_bounds__(256)
void gemm_wmma_f32(const float* __restrict__ A, const float* __restrict__ B,
                   const float* __restrict__ bias,
                   const unsigned char* __restrict__ rowmask,
                   float* __restrict__ C, int M, int N, int K)
{
    __shared__ float Bs[32][65];

    const int lane   = threadIdx.x & 31;
    const int wave   = threadIdx.x >> 5;
    const int m0     = blockIdx.x * 128 + wave * 16;
    const int nblk   = blockIdx.y * 64;
    const int half   = lane >> 4;          // 0: lanes 0-15, 1: lanes 16-31
    const int l16    = lane & 15;

    v8f acc0 = {0,0,0,0,0,0,0,0};
    v8f acc1 = acc0, acc2 = acc0, acc3 = acc0;

    int arow = m0 + l16; if (arow > M - 1) arow = M - 1;   // clamp, store-guarded
    const float* Aptr = A + (size_t)arow * K + 2 * half;

    for (int kc = 0; kc < K; kc += 32) {
        // cooperative stage of B[kc..kc+32) x [nblk..nblk+64) into LDS
        #pragma unroll
        for (int i = 0; i < 8; ++i) {
            int idx = threadIdx.x * 8 + i;         // 0..2047
            int kr  = idx >> 6, nr = idx & 63;
            int col = nblk + nr;
            Bs[kr][nr] = (col < N) ? B[(size_t)(kc + kr) * N + col] : 0.0f;
        }
        if (((threadIdx.x & 31) == 0) && (kc + 32 < K))
            __builtin_prefetch(&B[(size_t)(kc + 32) * N + nblk], 0, 1);
        __syncthreads();

        #pragma unroll
        for (int ks = 0; ks < 8; ++ks) {
            const float2 a2 = *(const float2*)(Aptr + kc + ks * 4);
            v2f af; af.x = a2.x; af.y = a2.y;
            const int kb = ks * 4 + 2 * half;
            v2f bf;
            bf.x = Bs[kb][ 0 + l16]; bf.y = Bs[kb + 1][ 0 + l16];
            acc0 = __builtin_amdgcn_wmma_f32_16x16x4_f32(false, af, false, bf, (short)0, acc0, false, false);
            bf.x = Bs[kb][16 + l16]; bf.y = Bs[kb + 1][16 + l16];
            acc1 = __builtin_amdgcn_wmma_f32_16x16x4_f32(false, af, false, bf, (short)0, acc1, false, false);
            bf.x = Bs[kb][32 + l16]; bf.y = Bs[kb + 1][32 + l16];
            acc2 = __builtin_amdgcn_wmma_f32_16x16x4_f32(false, af, false, bf, (short)0, acc2, false, false);
            bf.x = Bs[kb][48 + l16]; bf.y = Bs[kb + 1][48 + l16];
            acc3 = __builtin_amdgcn_wmma_f32_16x16x4_f32(false, af, false, bf, (short)0, acc3, false, false);
        }
        __syncthreads();
    }

    // epilogue: C/D layout -> lane<16: rows m0+0..7, lane>=16: rows m0+8..15
    v8f accs[4] = {acc0, acc1, acc2, acc3};
    #pragma unroll
    for (int s = 0; s < 4; ++s) {
        const int col = nblk + s * 16 + l16;
        if (col >= N) continue;
        const float bv = bias ? bias[col] : 0.0f;
        #pragma unroll
        for (int r = 0; r < 8; ++r) {
            const int row = m0 + half * 8 + r;
            if (row < M) {
                float v = accs[s][r] + bv;
                if (USE_MASK) { if (rowmask[row]) v = 0.0f; }
                if (EPI == EPI_GELU)
                    v = 0.5f * v * (1.0f + erff(v * 0.70710678118654752f));
                C[(size_t)row * N + col] = v;
            }
        }
    }
}

// ---------------------------------------------------------------------------
// LayerNorm over C=256, optional residual: Y = LN(X (+ R)) * g + b
// one wave32 per row, lane owns 8 channels, shfl_xor reduction.
// ---------------------------------------------------------------------------
__global__ __launch_bounds__(256)
void ln256(const float* __restrict__ X, const float* __restrict__ R,
           const float* __restrict__ g, const float* __restrict__ b,
           float* __restrict__ Y, int rows)
{
    const int lane = threadIdx.x & 31;
    const int wave = threadIdx.x >> 5;
    const int row  = blockIdx.x * 8 + wave;
    if (row >= rows) return;

    const float* x = X + (size_t)row * CDIM_;
    float v[8];
    #pragma unroll
    for (int i = 0; i < 8; ++i) v[i] = x[lane * 8 + i];
    if (R) {
        const float* rr = R + (size_t)row * CDIM_;
        #pragma unroll
        for (int i = 0; i < 8; ++i) v[i] += rr[lane * 8 + i];
    }
    float s = 0.f, s2 = 0.f;
    #pragma unroll
    for (int i = 0; i < 8; ++i) { s += v[i]; s2 += v[i] * v[i]; }
    #pragma unroll
    for (int off = 16; off > 0; off >>= 1) {
        s  += __shfl_xor(s,  off, 32);
        s2 += __shfl_xor(s2, off, 32);
    }
    const float mean = s * (1.0f / CDIM_);
    const float var  = s2 * (1.0f / CDIM_) - mean * mean;
    const float rstd = rsqrtf(var + 1e-5f);
    #pragma unroll
    for (int i = 0; i < 8; ++i) {
        const int c = lane * 8 + i;
        Y[(size_t)row * CDIM_ + c] = (v[i] - mean) * rstd * g[c] + b[c];
    }
}

// ---------------------------------------------------------------------------
// q = tgt + query_pos (float4)
// ---------------------------------------------------------------------------
__global__ __launch_bounds__(256)
void add_vec4(const float* __restrict__ a, const float* __restrict__ b,
              float* __restrict__ c, long n4)
{
    long i = (long)blockIdx.x * blockDim.x + threadIdx.x;
    if (i < n4) {
        float4 av = ((const float4*)a)[i];
        float4 bv = ((const float4*)b)[i];
        float4 cv = {av.x + bv.x, av.y + bv.y, av.z + bv.z, av.w + bv.w};
        ((float4*)c)[i] = cv;
    }
}

// ---------------------------------------------------------------------------
// Fused softmax + location + bilinear sampling + point reduction.
// one wave32 per (row, head); lane = channel d.  value is [B, LS, 8, 32]
// (natural [B,LS,256] layout) -> 128B coalesced gathers per corner from L2.
// ---------------------------------------------------------------------------
__global__ __launch_bounds__(256)
void deform_sample(const float* __restrict__ value,   // [B*LS, 256]
                   const float* __restrict__ offs,    // [rows, 64]
                   const float* __restrict__ logits,  // [rows, 32]
                   const float* __restrict__ refp,    // [B, LQ, 1, 2]
                   float* __restrict__ out,           // [rows, 256]
                   int rows)
{
    const int lane = threadIdx.x & 31;
    const int wave = threadIdx.x >> 5;
    const int gid  = blockIdx.x * 8 + wave;
    if (gid >= rows * 8) return;
    const int row  = gid >> 3;
    const int head = gid & 7;
    const int bidx = row / LQ_;

    // softmax over the L*P = 4 logits of this head
    float lg[4];
    #pragma unroll
    for (int p = 0; p < 4; ++p) lg[p] = logits[(size_t)row * 32 + head * 4 + p];
    float mx = fmaxf(fmaxf(lg[0], lg[1]), fmaxf(lg[2], lg[3]));
    float w[4], sum = 0.f;
    #pragma unroll
    for (int p = 0; p < 4; ++p) { w[p] = expf(lg[p] - mx); sum += w[p]; }
    const float inv = 1.0f / sum;
    #pragma unroll
    for (int p = 0; p < 4; ++p) w[p] *= inv;

    const float rx = refp[(size_t)row * 2 + 0];
    const float ry = refp[(size_t)row * 2 + 1];
    const float* vbase = value + (size_t)bidx * LS_ * CDIM_ + head * 32 + lane;

    float acc = 0.f;
    #pragma unroll
    for (int p = 0; p < 4; ++p) {
        const float ox = offs[(size_t)row * 64 + head * 8 + p * 2 + 0];
        const float oy = offs[(size_t)row * 64 + head * 8 + p * 2 + 1];
        // loc = ref + off/(W,H);  x = loc_x*W - 0.5 => rx*W + ox - 0.5
        const float x = rx * (float)WLVL_ + ox - 0.5f;
        const float y = ry * (float)HLVL_ + oy - 0.5f;
        const float x0f = floorf(x), y0f = floorf(y);
        const float lx = x - x0f, ly = y - y0f;
        const int x0 = (int)x0f, y0 = (int)y0f;

        #pragma unroll
        for (int cy = 0; cy < 2; ++cy) {
            #pragma unroll
            for (int cx = 0; cx < 2; ++cx) {
                const int xi = x0 + cx, yi = y0 + cy;
                const bool valid = (xi >= 0) & (xi < WLVL_) & (yi >= 0) & (yi < HLVL_);
                int xc = xi < 0 ? 0 : (xi > WLVL_ - 1 ? WLVL_ - 1 : xi);
                int yc = yi < 0 ? 0 : (yi > HLVL_ - 1 ? HLVL_ - 1 : yi);
                const float v = vbase[(size_t)(yc * WLVL_ + xc) * CDIM_];
                const float ww = (cx ? lx : 1.f - lx) * (cy ? ly : 1.f - ly);
                acc += (valid ? v : 0.f) * ww * w[p];
            }
        }
    }
    out[(size_t)row * CDIM_ + head * 32 + lane] = acc;
}

// ---------------------------------------------------------------------------
extern "C" void kernel_launch(void* const* d_in, const int* in_sizes, int n_in,
                              void* d_out, int out_size, void* d_ws, size_t ws_size,
                              hipStream_t stream)
{
    const float* tgt    = (const float*)d_in[0];
    const float* qpos   = (const float*)d_in[1];
    const float* refp   = (const float*)d_in[4];
    const float* src    = (const float*)d_in[6];
    const unsigned char* pmask = (const unsigned char*)d_in[9];
    const float* W_dsa  = (const float*)d_in[10]; const float* b_dsa  = (const float*)d_in[11];
    const float* g_nds  = (const float*)d_in[12]; const float* b_nds  = (const float*)d_in[13];
    const float* W_off  = (const float*)d_in[14]; const float* b_off  = (const float*)d_in[15];
    const float* W_attn = (const float*)d_in[16]; const float* b_attn = (const float*)d_in[17];
    const float* W_val  = (const float*)d_in[18]; const float* b_val  = (const float*)d_in[19];
    const float* W_out  = (const float*)d_in[20]; const float* b_out  = (const float*)d_in[21];
    const float* W_csa  = (const float*)d_in[22]; const float* b_csa  = (const float*)d_in[23];
    const float* g_n1   = (const float*)d_in[24]; const float* b_n1   = (const float*)d_in[25];
    const float* W_ff   = (const float*)d_in[26]; const float* b_ff   = (const float*)d_in[27];
    const float* g_n3   = (const float*)d_in[28]; const float* b_n3   = (const float*)d_in[29];
    float* outp = (float*)d_out;

    const int M = ROWS_, C = CDIM_;
    const size_t S = (size_t)M * C;

    // rotating scratch: R0 = q / attn_out / t ; R1 = dsa_tmp / value / tgt2 ;
    // R2 = src_a / tgt2a / ffh ; then offsets + logits.  Total ~104 MB.
    float* R0   = (float*)d_ws;
    float* R1   = R0 + S;
    float* R2   = R1 + S;
    float* OFFS = R2 + S;                       // [M,64]
    float* LOGI = OFFS + (size_t)M * 64;        // [M,32]

    const dim3 blk(256);
    const dim3 gM128N256((M + 127) / 128, (C + 63) / 64);
    const dim3 gM128N64((M + 127) / 128, 1);
    const dim3 gLN((M + 7) / 8);

    // 1. q = tgt + query_pos
    const long n4 = (long)(S / 4);
    add_vec4<<<dim3((unsigned)((n4 + 255) / 256)), blk, 0, stream>>>(tgt, qpos, R0, n4);

    // 2. depth-scale adaptation: src_a = LN(src @ W_dsa + b_dsa)
    gemm_wmma_f32<EPI_NONE, false><<<gM128N256, blk, 0, stream>>>(src, W_dsa, b_dsa, nullptr, R1, M, C, C);
    ln256<<<gLN, blk, 0, stream>>>(R1, nullptr, g_nds, b_nds, R2, M);

    // 3. value = (src_a @ W_val + b_val) masked by padding
    gemm_wmma_f32<EPI_NONE, true><<<gM128N256, blk, 0, stream>>>(R2, W_val, b_val, pmask, R1, M, C, C);

    // 4. offsets and attention logits from q
    gemm_wmma_f32<EPI_NONE, false><<<gM128N64, blk, 0, stream>>>(R0, W_off,  b_off,  nullptr, OFFS, M, 64, C);
    gemm_wmma_f32<EPI_NONE, false><<<gM128N64, blk, 0, stream>>>(R0, W_attn, b_attn, nullptr, LOGI, M, 32, C);

    // 5. softmax + bilinear deformable sampling -> attn_out (reuses q buffer)
    deform_sample<<<dim3((unsigned)M), blk, 0, stream>>>(R1, OFFS, LOGI, refp, R0, M);

    // 6. output proj + cross-scale proj
    gemm_wmma_f32<EPI_NONE, false><<<gM128N256, blk, 0, stream>>>(R0, W_out, b_out, nullptr, R2, M, C, C);
    gemm_wmma_f32<EPI_NONE, false><<<gM128N256, blk, 0, stream>>>(R2, W_csa, b_csa, nullptr, R1, M, C, C);

    // 7. t = LN(tgt + tgt2)
    ln256<<<gLN, blk, 0, stream>>>(R1, tgt, g_n1, b_n1, R0, M);

    // 8. ff = GELU(t @ W_ff + b_ff)   (exact erf GELU)
    gemm_wmma_f32<EPI_GELU, false><<<gM128N256, blk, 0, stream>>>(R0, W_ff, b_ff, nullptr, R2, M, C, C);

    // 9. out = LN(t + ff)
    ln256<<<gLN, blk, 0, stream>>>(R2, R0, g_n3, b_n3, outp, M);
}